// SpectralSGCN2Layer_67585605369887
// MI455X (gfx1250) — compile-verified
//
#include <hip/hip_runtime.h>
#include <hip/hip_bf16.h>

// SpectralSGCN2Layer for MI455X (gfx1250, wave32).
//
// Phase 1: z = 0                                  (float4 stores, 51.2 MB)
// Phase 2: [s_dst|s_src] = h @ [w_dst w_src 0..]  via V_WMMA_F32_16X16X4_F32
// Phase 3: per edge e: coeff = tanh(s_dst[dst]+s_src[src]+b)*d[dst]*d[src]*w[e]
//          z[dst] += coeff * h[src]               (wave/edge, f32 L2 atomics)
//
// Roofline: FLOPs ~0.41 GFLOP (negligible). Traffic: edge phase moves
// E*512B gather + E*512B atomic-add = ~1.6 GB, but h (51.2MB) and z (51.2MB)
// both fit in the 192MB L2 -> L2-bandwidth bound, atomics non-returning.

typedef __attribute__((ext_vector_type(2))) float v2f;
typedef __attribute__((ext_vector_type(8))) float v8f;

#define D 128

__global__ void zero_kernel(float4* __restrict__ z, int n4) {
    int i = blockIdx.x * blockDim.x + threadIdx.x;
    if (i < n4) z[i] = make_float4(0.f, 0.f, 0.f, 0.f);
}

// One wave (32 threads) per 16-row tile of h. Computes D = A(16x128) x B(128x16)
// as 32 chained V_WMMA_F32_16X16X4_F32 ops. B columns: 0 = w_dst, 1 = w_src,
// 2..15 = zero. The B fragment is built with ZERO control flow in the loop:
// a lane-varying base pointer chosen once (v_cndmask on the pointer) plus a
// loop-invariant multiplicative mask (lowers to v_pk_mul_f32). EXEC stays
// all-ones through every WMMA; loads cannot be sunk into predicated blocks.
__global__ void gate_proj_wmma(const float* __restrict__ h,
                               const float* __restrict__ gate_w,
                               float* __restrict__ s_dst,
                               float* __restrict__ s_src,
                               int N) {
    const int lane  = threadIdx.x;        // 0..31, one full wave per block
    const int tile  = blockIdx.x;         // 16 rows per tile
    const int m     = lane & 15;          // row within tile (A: M = lane%16)
    const int khalf = (lane >> 4) * 2;    // lanes 16-31 hold K = k0+2, k0+3

    int row = tile * 16 + m;
    if (row >= N) row = N - 1;            // tail: duplicate loads, writes guarded

    // Column m of B reads from: m==0 -> w_dst, m==1 -> w_src, else zeroed.
    const float* __restrict__ bptr   = (m == 1) ? (gate_w + D) : gate_w;
    const float               bscale = (m <= 1) ? 1.0f : 0.0f;
    const float* __restrict__ hrow   = h + (long long)row * D;

    v8f c = {};
    #pragma unroll 4
    for (int k0 = 0; k0 < D; k0 += 4) {
        const int kk = k0 + khalf;
        // A-matrix 16x4 f32 layout: VGPR0 = K{0|2}, VGPR1 = K{1|3}
        const float2 av = *(const float2*)(hrow + kk);
        // Single unconditional (half-wave-broadcast, L1-hot) weight load.
        const float2 bv = *(const float2*)(bptr + kk);
        v2f a, b;
        a.x = av.x;
        a.y = av.y;
        b.x = bv.x * bscale;
        b.y = bv.y * bscale;
        c = __builtin_amdgcn_wmma_f32_16x16x4_f32(
                /*neg_a=*/false, a, /*neg_b=*/false, b,
                /*c_mod=*/(short)0, c, /*reuse_a=*/false, /*reuse_b=*/false);
    }

    // Extract columns N=0 (s_dst) and N=1 (s_src) from the 16x16 f32 C/D
    // layout: VGPR r -> lanes0-15 (M=r, N=lane), lanes16-31 (M=8+r, N=lane-16).
    // base is 8-aligned -> 32B-aligned address -> two b128 stores on fast path.
    if (m <= 1) {
        float* __restrict__ out = (m == 0) ? s_dst : s_src;
        const int base = tile * 16 + ((lane >> 4) ? 8 : 0);  // lanes 16/17 -> M=8..15
        if (base + 8 <= N) {
            float4 lo = make_float4(c[0], c[1], c[2], c[3]);
            float4 hi = make_float4(c[4], c[5], c[6], c[7]);
            *(float4*)(out + base)     = lo;
            *(float4*)(out + base + 4) = hi;
        } else {
            #pragma unroll
            for (int r = 0; r < 8; ++r) {
                if (base + r < N) out[base + r] = c[r];
            }
        }
    }
}

// One wave per edge (grid-stride). 32 lanes x float4 = one full 512B h-row per
// request. Edge scalars are wave-uniform loads (coalesce to one request each).
__global__ void edge_scatter(const float* __restrict__ h,
                             const float* __restrict__ dvec,
                             const float* __restrict__ w,
                             const float* __restrict__ gate_b,
                             const int*   __restrict__ src,
                             const int*   __restrict__ dst,
                             const float* __restrict__ s_dst,
                             const float* __restrict__ s_src,
                             float* __restrict__ z,
                             int E, int total_waves) {
    const int wave = (blockIdx.x * blockDim.x + threadIdx.x) >> 5;
    const int lane = threadIdx.x & 31;
    const float gb = gate_b[0];

    for (int e = wave; e < E; e += total_waves) {
        const int se = src[e];
        const int de = dst[e];
        const float alpha = tanhf(s_dst[de] + s_src[se] + gb);
        const float coeff = alpha * dvec[de] * dvec[se] * w[e];

        const float4 v = ((const float4*)(h + (long long)se * D))[lane];
        float* __restrict__ zrow = z + (long long)de * D + lane * 4;
        // Non-returning f32 atomics -> global_atomic_add_f32, resolved in L2.
        atomicAdd(zrow + 0, v.x * coeff);
        atomicAdd(zrow + 1, v.y * coeff);
        atomicAdd(zrow + 2, v.z * coeff);
        atomicAdd(zrow + 3, v.w * coeff);
    }
}

extern "C" void kernel_launch(void* const* d_in, const int* in_sizes, int n_in,
                              void* d_out, int out_size, void* d_ws, size_t ws_size,
                              hipStream_t stream) {
    const float* h      = (const float*)d_in[0];
    const float* dvec   = (const float*)d_in[1];
    const float* w      = (const float*)d_in[2];
    const float* gate_w = (const float*)d_in[3];
    const float* gate_b = (const float*)d_in[4];
    const int*   src    = (const int*)d_in[5];
    const int*   dst    = (const int*)d_in[6];
    float* z = (float*)d_out;

    const int N = in_sizes[0] / D;   // 100000
    const int E = in_sizes[2];       // 1600000

    // Workspace: s_dst[N], s_src[N]  (~800 KB)
    float* s_dst = (float*)d_ws;
    float* s_src = s_dst + N;

    // 1) z = 0
    const int n4 = N * (D / 4);
    zero_kernel<<<(n4 + 255) / 256, 256, 0, stream>>>((float4*)z, n4);

    // 2) gate projections via WMMA, one wave per 16-row tile
    gate_proj_wmma<<<(N + 15) / 16, 32, 0, stream>>>(h, gate_w, s_dst, s_src, N);

    // 3) edge gather/scale/scatter, one wave per edge, grid-stride
    int blocks = 4096;                      // 32768 waves, ~49 edges each
    const int blk_need = (E + 7) / 8;       // 8 waves per 256-thread block
    if (blk_need < blocks) blocks = blk_need;
    edge_scatter<<<blocks, 256, 0, stream>>>(h, dvec, w, gate_b, src, dst,
                                             s_dst, s_src, z, E, blocks * 8);
}